// MeanAggregator_2018634629566
// MI455X (gfx1250) — compile-verified
//
#include <hip/hip_runtime.h>

// CDNA5 / gfx1250, wave32.
typedef __attribute__((ext_vector_type(2))) float v2f;
typedef __attribute__((ext_vector_type(8))) float v8f;

#define FEAT_DIM 128
#define KNEIGH   16
#define WAVES_PER_BLOCK 8   // 256 threads = 8 wave32

// One wave per output row. The K=16 mean reduction is executed on the matrix
// pipe: D = A(16x4, all 1/16) x B(4x16 gathered neighbor columns) + C,
// accumulated over 4 chunks of 4 neighbors. Every row of D holds the mean for
// 16 feature columns; lanes 0-15 of VGPR0 (acc element 0) are stored.
__global__ __launch_bounds__(256) void mean_agg_wmma_kernel(
    const float* __restrict__ feat,   // [NUM_NODES, 128]
    const int*   __restrict__ nidx,   // [batch, 16]
    float*       __restrict__ out,    // [batch, 128]
    int batch)
{
    const int lane = threadIdx.x & 31;
    const int wave = threadIdx.x >> 5;
    const int row  = blockIdx.x * WAVES_PER_BLOCK + wave;
    if (row >= batch) return;          // wave-uniform: EXEC stays all-1s below

    const int* __restrict__ ridx = nidx + (size_t)row * KNEIGH;
    const int half = lane >> 4;        // 0: neighbors 4c+0/4c+1, 1: 4c+2/4c+3
    const int col  = lane & 15;        // feature column within group (== N lane map)

    // Hoist the two neighbor-row base pointers this lane uses in each chunk,
    // and prefetch the rows so the first group's reads hit warm lines.
    const float* base0[4];
    const float* base1[4];
#pragma unroll
    for (int c = 0; c < 4; ++c) {
        const int i0 = ridx[c * 4 + half * 2 + 0];
        const int i1 = ridx[c * 4 + half * 2 + 1];
        base0[c] = feat + ((size_t)i0 << 7);   // *FEAT_DIM
        base1[c] = feat + ((size_t)i1 << 7);
        __builtin_prefetch(base0[c], 0, 3);    // -> global_prefetch_b8
        __builtin_prefetch(base1[c], 0, 3);
    }

    // A-matrix: 16x4 of (1/16), identical in every lane/VGPR slot.
    const v2f a = {0.0625f, 0.0625f};

    float* __restrict__ orow = out + (size_t)row * FEAT_DIM;

#pragma unroll
    for (int g = 0; g < 8; ++g) {
        const int o = g * 16 + col;
        v8f acc = {};
#pragma unroll
        for (int c = 0; c < 4; ++c) {
            v2f b;
            b.x = base0[c][o];   // B row j   , column (L&15)
            b.y = base1[c][o];   // B row j+1 , column (L&15)
            // 8 args: (neg_a, A, neg_b, B, c_mod, C, reuse_a, reuse_b)
            acc = __builtin_amdgcn_wmma_f32_16x16x4_f32(
                false, a, false, b, (short)0, acc, false, false);
        }
        // D layout: VGPR0, lanes 0-15 hold M=0, N=0..15 (all D rows are equal).
        if (half == 0) {
            __builtin_nontemporal_store(acc[0], orow + o);  // write-once output
        }
    }
}

extern "C" void kernel_launch(void* const* d_in, const int* in_sizes, int n_in,
                              void* d_out, int out_size, void* d_ws, size_t ws_size,
                              hipStream_t stream) {
    const float* feat = (const float*)d_in[0];   // features [NUM_NODES,128] f32
    const int*   nidx = (const int*)d_in[1];     // neigh_idx [batch,16] int
    float*       out  = (float*)d_out;           // [batch,128] f32

    const int batch  = in_sizes[1] / KNEIGH;
    const int blocks = (batch + WAVES_PER_BLOCK - 1) / WAVES_PER_BLOCK;

    mean_agg_wmma_kernel<<<blocks, 256, 0, stream>>>(feat, nidx, out, batch);
}